// KWinner_61194694034264
// MI455X (gfx1250) — compile-verified
//
#include <hip/hip_runtime.h>

#define UNITS   131072
#define BATCH   256
#define KSEL    13107u
#define ALPHA_F (1.0f / 1000.0f)
#define TILE    8192
#define NTILES  (UNITS / TILE)
#define NTHREAD 1024
#define CHUNK   (UNITS / NTHREAD)   // 128 contiguous elements per thread in pass D
#define HCOPY   4                   // privatized histogram copies (contention relief)

typedef __attribute__((ext_vector_type(4))) unsigned int u32x4;
typedef __attribute__((ext_vector_type(4))) int          i32x4;
typedef __attribute__((ext_vector_type(8))) int          i32x8;

// float -> order-preserving unsigned key (larger float => larger key)
__device__ __forceinline__ unsigned key_of(float f) {
    unsigned u = __float_as_uint(f);
    return (u & 0x80000000u) ? ~u : (u | 0x80000000u);
}

__device__ __forceinline__ void wait_tensor0() {
#if __has_builtin(__builtin_amdgcn_s_wait_tensorcnt)
    __builtin_amdgcn_s_wait_tensorcnt(0);
#else
    asm volatile("s_wait_tensorcnt 0x0" ::: "memory");
#endif
}

// TDM: DMA one contiguous tile of TILE floats into LDS (CDNA5 ISA ch.8 descriptor).
__device__ __forceinline__ void tdm_load_tile(const float* gsrc, unsigned lds_byte_off) {
    unsigned long long ga = (unsigned long long)gsrc;
    u32x4 g0;
    g0[0] = 1u;                                                   // count=1, valid
    g0[1] = lds_byte_off;                                         // lds_addr
    g0[2] = (unsigned)(ga & 0xFFFFFFFFull);                       // global_addr[31:0]
    g0[3] = (unsigned)((ga >> 32) & 0x01FFFFFFull) | 0x80000000u; // addr[56:32] | type=2
    i32x8 g1;
    g1[0] = (2 << 16);                 // workgroup_mask=0, data_size=2 (4 bytes)
    g1[1] = 0;                         // tensor_dim0[15:0]=0 (131072 lo)
    g1[2] = 2 | (1 << 16);             // tensor_dim0[31:16]=2, tensor_dim1=1
    g1[3] = (TILE << 16);              // tile_dim0=TILE
    g1[4] = 1;                         // tile_dim1=1
    g1[5] = UNITS;                     // tensor_dim0_stride lo
    g1[6] = 0;
    g1[7] = 0;
    i32x4 z4 = {0, 0, 0, 0};
#if __has_builtin(__builtin_amdgcn_tensor_load_to_lds)
#if __clang_major__ >= 23
    i32x8 z8 = {0, 0, 0, 0, 0, 0, 0, 0};
    __builtin_amdgcn_tensor_load_to_lds(g0, g1, z4, z4, z8, 0);
#else
    __builtin_amdgcn_tensor_load_to_lds(g0, g1, z4, z4, 0);
#endif
#else
    asm volatile("tensor_load_to_lds %0, %1" :: "s"(g0), "s"(g1) : "memory");
#endif
}

// ---------------- kernel 1: boost + zero counters ----------------
__global__ __launch_bounds__(256) void kwta_prep(const float* __restrict__ duty,
                                                 float* __restrict__ boost,
                                                 unsigned* __restrict__ counts) {
    int u = blockIdx.x * 256 + threadIdx.x;
    const float target = (float)KSEL / (float)UNITS;
    boost[u]  = expf(target - duty[u]);   // BETA = 1
    counts[u] = 0u;
}

// ---------------- kernel 2: per-row radix-select + output ----------------
__global__ __launch_bounds__(NTHREAD) void kwta_row(const float* __restrict__ in,
                                                    const float* __restrict__ boost,
                                                    float* __restrict__ out,
                                                    unsigned* __restrict__ counts) {
    __shared__ unsigned hist[HCOPY][4096];
    __shared__ float    tile[2][TILE];
    __shared__ unsigned eqarr[NTHREAD];
    __shared__ unsigned sbin, sabove;

    const int tid  = threadIdx.x;
    const int copy = (tid >> 8) & (HCOPY - 1);   // 8 waves per histogram copy
    const int row  = blockIdx.x;
    const float* rowp = in + (size_t)row * UNITS;
    float*       outp = out + (size_t)row * UNITS;
    const float4* r4 = reinterpret_cast<const float4*>(rowp);
    const float4* b4 = reinterpret_cast<const float4*>(boost);

    unsigned lds_off[2];
    lds_off[0] = (unsigned)(unsigned long long)(void*)&tile[0][0];
    lds_off[1] = (unsigned)(unsigned long long)(void*)&tile[1][0];

    // ================= Pass A: 12-bit histogram via TDM-staged tiles =================
    #pragma unroll
    for (int c = 0; c < HCOPY; ++c)
        #pragma unroll
        for (int j = 0; j < 4; ++j) hist[c][tid + j * NTHREAD] = 0u;
    if (tid < 32) {                       // wave 0 drives the DMA
        tdm_load_tile(rowp, lds_off[0]);
        wait_tensor0();
    }
    __syncthreads();

    for (int t = 0; t < NTILES; ++t) {
        int cur = t & 1;
        if (tid < 32 && (t + 1) < NTILES)
            tdm_load_tile(rowp + (size_t)(t + 1) * TILE, lds_off[cur ^ 1]);
        #pragma unroll
        for (int j = 0; j < TILE / NTHREAD; ++j) {
            int idx = tid + j * NTHREAD;
            int u   = t * TILE + idx;
            float x = tile[cur][idx];
            unsigned k = key_of(x * boost[u]);
            atomicAdd(&hist[copy][k >> 20], 1u);
        }
        if (tid < 32 && (t + 1) < NTILES) wait_tensor0();
        __syncthreads();
    }

    // merge copies -> hist[0], inclusive suffix scan, find bin
    #pragma unroll
    for (int j = 0; j < 4; ++j) {
        int i = tid + j * NTHREAD;
        hist[0][i] += hist[1][i] + hist[2][i] + hist[3][i];
    }
    __syncthreads();
    for (int off = 1; off < 4096; off <<= 1) {
        unsigned v[4];
        #pragma unroll
        for (int j = 0; j < 4; ++j) {
            int i = tid + j * NTHREAD;
            unsigned a = hist[0][i];
            unsigned b = (i + off < 4096) ? hist[0][i + off] : 0u;
            v[j] = a + b;
        }
        __syncthreads();
        #pragma unroll
        for (int j = 0; j < 4; ++j) hist[0][tid + j * NTHREAD] = v[j];
        __syncthreads();
    }
    #pragma unroll
    for (int j = 0; j < 4; ++j) {
        int i = tid + j * NTHREAD;
        unsigned si = hist[0][i];
        unsigned sn = (i < 4095) ? hist[0][i + 1] : 0u;
        if (si >= KSEL && sn < KSEL) { sbin = (unsigned)i; sabove = sn; }
    }
    __syncthreads();
    unsigned b1 = sbin;
    unsigned R1 = KSEL - sabove;
    __syncthreads();

    // ================= Pass B: refine middle 12 bits (L2-resident reread) =================
    #pragma unroll
    for (int c = 0; c < HCOPY; ++c)
        #pragma unroll
        for (int j = 0; j < 4; ++j) hist[c][tid + j * NTHREAD] = 0u;
    __syncthreads();
    for (int c = 0; c < UNITS / 4 / NTHREAD; ++c) {
        int v = c * NTHREAD + tid;
        float4 x = r4[v], bb = b4[v];
        const float* xe = (const float*)&x;
        const float* be = (const float*)&bb;
        #pragma unroll
        for (int e = 0; e < 4; ++e) {
            unsigned k = key_of(xe[e] * be[e]);
            if ((k >> 20) == b1) atomicAdd(&hist[copy][(k >> 8) & 0xFFFu], 1u);
        }
    }
    __syncthreads();
    #pragma unroll
    for (int j = 0; j < 4; ++j) {
        int i = tid + j * NTHREAD;
        hist[0][i] += hist[1][i] + hist[2][i] + hist[3][i];
    }
    __syncthreads();
    for (int off = 1; off < 4096; off <<= 1) {
        unsigned v[4];
        #pragma unroll
        for (int j = 0; j < 4; ++j) {
            int i = tid + j * NTHREAD;
            unsigned a = hist[0][i];
            unsigned b = (i + off < 4096) ? hist[0][i + off] : 0u;
            v[j] = a + b;
        }
        __syncthreads();
        #pragma unroll
        for (int j = 0; j < 4; ++j) hist[0][tid + j * NTHREAD] = v[j];
        __syncthreads();
    }
    #pragma unroll
    for (int j = 0; j < 4; ++j) {
        int i = tid + j * NTHREAD;
        unsigned si = hist[0][i];
        unsigned sn = (i < 4095) ? hist[0][i + 1] : 0u;
        if (si >= R1 && sn < R1) { sbin = (unsigned)i; sabove = sn; }
    }
    __syncthreads();
    unsigned b2 = sbin;
    unsigned R2 = R1 - sabove;
    __syncthreads();

    // ================= Pass C: last 8 bits =================
    #pragma unroll
    for (int c = 0; c < HCOPY; ++c)
        #pragma unroll
        for (int j = 0; j < 4; ++j) hist[c][tid + j * NTHREAD] = 0u;
    __syncthreads();
    unsigned pref24 = (b1 << 12) | b2;
    for (int c = 0; c < UNITS / 4 / NTHREAD; ++c) {
        int v = c * NTHREAD + tid;
        float4 x = r4[v], bb = b4[v];
        const float* xe = (const float*)&x;
        const float* be = (const float*)&bb;
        #pragma unroll
        for (int e = 0; e < 4; ++e) {
            unsigned k = key_of(xe[e] * be[e]);
            if ((k >> 8) == pref24) atomicAdd(&hist[copy][k & 0xFFu], 1u);
        }
    }
    __syncthreads();
    #pragma unroll
    for (int j = 0; j < 4; ++j) {
        int i = tid + j * NTHREAD;
        hist[0][i] += hist[1][i] + hist[2][i] + hist[3][i];
    }
    __syncthreads();
    for (int off = 1; off < 4096; off <<= 1) {   // bins 256..4095 are zero; harmless
        unsigned v[4];
        #pragma unroll
        for (int j = 0; j < 4; ++j) {
            int i = tid + j * NTHREAD;
            unsigned a = hist[0][i];
            unsigned b = (i + off < 4096) ? hist[0][i + off] : 0u;
            v[j] = a + b;
        }
        __syncthreads();
        #pragma unroll
        for (int j = 0; j < 4; ++j) hist[0][tid + j * NTHREAD] = v[j];
        __syncthreads();
    }
    #pragma unroll
    for (int j = 0; j < 4; ++j) {
        int i = tid + j * NTHREAD;
        unsigned si = hist[0][i];
        unsigned sn = (i < 4095) ? hist[0][i + 1] : 0u;
        if (si >= R2 && sn < R2) { sbin = (unsigned)i; sabove = sn; }
    }
    __syncthreads();
    unsigned b3     = sbin;
    unsigned needEq = R2 - sabove;                 // #ties at T to keep (lowest index first)
    unsigned T      = (b1 << 20) | (b2 << 8) | b3; // exact threshold key
    __syncthreads();

    // ========== Pass D: contiguous chunks, exact index-order tie-break, write out ==========
    unsigned eq = 0;
    #pragma unroll 4
    for (int j = 0; j < CHUNK / 4; ++j) {
        int v = tid * (CHUNK / 4) + j;
        float4 x = r4[v], bb = b4[v];
        const float* xe = (const float*)&x;
        const float* be = (const float*)&bb;
        #pragma unroll
        for (int e = 0; e < 4; ++e)
            eq += (key_of(xe[e] * be[e]) == T) ? 1u : 0u;
    }
    eqarr[tid] = eq;
    __syncthreads();
    for (int off = 1; off < NTHREAD; off <<= 1) {
        unsigned v = eqarr[tid];
        if (tid >= off) v += eqarr[tid - off];
        __syncthreads();
        eqarr[tid] = v;
        __syncthreads();
    }
    unsigned run = eqarr[tid] - eq;                // exclusive prefix of equal-count

    float4* o4 = reinterpret_cast<float4*>(outp);
    #pragma unroll 4
    for (int j = 0; j < CHUNK / 4; ++j) {
        int v = tid * (CHUNK / 4) + j;
        float4 x = r4[v], bb = b4[v];
        const float* xe = (const float*)&x;
        const float* be = (const float*)&bb;
        float4 o;
        float* oe = (float*)&o;
        #pragma unroll
        for (int e = 0; e < 4; ++e) {
            unsigned k = key_of(xe[e] * be[e]);
            bool sel;
            if (k > T)       sel = true;
            else if (k == T) { sel = (run < needEq); run++; }
            else             sel = false;
            oe[e] = sel ? xe[e] : 0.0f;
            if (sel) atomicAdd(&counts[v * 4 + e], 1u);  // deterministic int atomics
        }
        o4[v] = o;
    }
}

// ---------------- kernel 3: EMA duty-cycle update ----------------
__global__ __launch_bounds__(256) void kwta_duty(const float* __restrict__ duty,
                                                 const unsigned* __restrict__ counts,
                                                 float* __restrict__ new_duty) {
    int u = blockIdx.x * 256 + threadIdx.x;
    float cur = (float)counts[u] * (1.0f / (float)BATCH);
    new_duty[u] = (1.0f - ALPHA_F) * duty[u] + ALPHA_F * cur;
}

extern "C" void kernel_launch(void* const* d_in, const int* in_sizes, int n_in,
                              void* d_out, int out_size, void* d_ws, size_t ws_size,
                              hipStream_t stream) {
    const float* inputs = (const float*)d_in[0];   // [BATCH, UNITS] f32
    const float* duty   = (const float*)d_in[1];   // [UNITS] f32
    float* out      = (float*)d_out;               // [BATCH*UNITS] out, then [UNITS] duty
    float* new_duty = out + (size_t)BATCH * UNITS;
    float*    boost  = (float*)d_ws;                                              // 512 KB
    unsigned* counts = (unsigned*)((char*)d_ws + (size_t)UNITS * sizeof(float));  // 512 KB

    kwta_prep<<<UNITS / 256, 256, 0, stream>>>(duty, boost, counts);
    kwta_row <<<BATCH, NTHREAD, 0, stream>>>(inputs, boost, out, counts);
    kwta_duty<<<UNITS / 256, 256, 0, stream>>>(duty, counts, new_duty);
}